// ROIAlign_63479616635498
// MI455X (gfx1250) — compile-verified
//
#include <hip/hip_runtime.h>

// ROI-align (N=1000, C=256, H=W=200, out 7x7) using V_WMMA_F32_16X16X4_F32
// One wave handles (roi, oh, 16-channel tile); 7 chained f32 WMMAs build the
// 16ch x 7ow tile. Feature map (164MB) is L2-resident (192MB L2).

#define SCALE   0.25f
#define NB      4
#define NC      256
#define NH      200
#define NW      200
#define NROI    1000
#define OH      7
#define OW      7
#define CT      16                 // channels per wave
#define NCT     (NC / CT)          // 16 tiles
#define NWAVES  (NROI * OH * NCT)  // 112000
#define WPB     8                  // waves per block
#define BLOCK   (WPB * 32)

typedef __attribute__((ext_vector_type(2))) float v2f;
typedef __attribute__((ext_vector_type(8))) float v8f;

// 8-byte pair load, only 4-byte alignment guaranteed
typedef struct __attribute__((packed, aligned(4))) { float x, y; } f2u;

__global__ __launch_bounds__(BLOCK) void roi_align_wmma_kernel(
    const float* __restrict__ fm, const float* __restrict__ rois,
    float* __restrict__ out)
{
    const int lane = threadIdx.x & 31;
    const int wave = blockIdx.x * WPB + (threadIdx.x >> 5);

    const int n   = wave / (OH * NCT);
    const int rem = wave % (OH * NCT);
    const int oh  = rem / NCT;
    const int ct  = rem % NCT;

    // ---- ROI params (wave-uniform; double-SCALE quirk kept faithful) ----
    const float c0 = rois[n * 5 + 0] * (SCALE * SCALE);
    const float c1 = rois[n * 5 + 1] * (SCALE * SCALE);
    const float c2 = rois[n * 5 + 2] * (SCALE * SCALE);
    const float c3 = rois[n * 5 + 3] * (SCALE * SCALE);
    int b = (int)(rois[n * 5 + 4] * SCALE);
    b = b < 0 ? 0 : (b > NB - 1 ? NB - 1 : b);

    const float x_min = fminf(fmaxf(c0, 0.f), (float)(NW - 1));
    const float y_min = fminf(fmaxf(c1, 0.f), (float)(NH - 1));
    const float x_max = fminf(fmaxf(c2, x_min + 1.f), (float)NW);
    const float y_max = fminf(fmaxf(c3, y_min + 1.f), (float)NH);
    const float sx = x_max - x_min, tx = 2.f * x_min / (float)NW - 1.f;
    const float sy = y_max - y_min, ty = 2.f * y_min / (float)NH - 1.f;

    // ---- y params for this oh; lane half selects y0 vs y1 row ----
    const float ysn = (2.f * (float)oh + 1.f) / (float)OH - 1.f;
    const float py  = ((sy * ysn + ty + 1.f) * (float)NH - 1.f) * 0.5f;
    const float y0f = floorf(py);
    const float wy1 = py - y0f;
    const float wy0 = 1.f - wy1;
    const int   y0i = (int)y0f;
    const int   y1i = y0i + 1;
    const float wy0v = (y0i >= 0 && y0i < NH) ? wy0 : 0.f;
    const float wy1v = (y1i >= 0 && y1i < NH) ? wy1 : 0.f;
    const int   yc0  = min(max(y0i, 0), NH - 1);
    const int   yc1  = min(max(y1i, 0), NH - 1);

    const bool  hi   = lane >= 16;
    const int   yrow = hi ? yc1 : yc0;
    const float wyv  = hi ? wy1v : wy0v;

    // ---- x params for all 7 output columns (lane-uniform) ----
    int   xbase[OW];
    int   selx0[OW], selx1[OW];      // branchless edge handling
    float wx0v[OW], wx1v[OW];
#pragma unroll
    for (int ow = 0; ow < OW; ++ow) {
        const float xsn = (2.f * (float)ow + 1.f) / (float)OW - 1.f;
        const float px  = ((sx * xsn + tx + 1.f) * (float)NW - 1.f) * 0.5f;
        const float x0f = floorf(px);
        const float w1  = px - x0f;
        const int   x0  = (int)x0f;
        const int   x1  = x0 + 1;
        wx0v[ow] = (x0 >= 0 && x0 < NW) ? (1.f - w1) : 0.f;
        wx1v[ow] = (x1 >= 0 && x1 < NW) ? w1 : 0.f;
        // load pair at xb, xb+1; xb clamped to [0, NW-2]
        const int xb = min(max(x0, 0), NW - 2);
        xbase[ow] = xb;
        selx0[ow] = (x0 == NW - 1) ? 1 : 0;  // x0 maps to .y when x0==W-1
        selx1[ow] = (x1 == 0)      ? 1 : 0;  // x1 maps to .x when x0==-1
    }

    // ---- per-lane gather base: channel m = lane&15, row per lane half ----
    const int m  = lane & 15;
    const int ch = ct * CT + m;
    const float* frow = fm + (((size_t)b * NC + ch) * NH + yrow) * NW;

    // ---- B weights: lane m supplies column ow==m (K halves split by lane half)
    const float bw0 = (m < OW) ? wyv * wx0v[m < OW ? m : 0] : 0.f; // wyv*wx0
    const float bw1 = (m < OW) ? wyv * wx1v[m < OW ? m : 0] : 0.f; // wyv*wx1

    v8f acc = {};
#pragma unroll
    for (int ow = 0; ow < OW; ++ow) {
        const f2u p = *(const f2u*)(frow + xbase[ow]);
        v2f a, bm;
        a.x = selx0[ow] ? p.y : p.x;   // value at x0 (clamped)
        a.y = selx1[ow] ? p.x : p.y;   // value at x1 (clamped)
        bm.x = (m == ow) ? bw0 : 0.f;  // B[K0/K2, ow] = wy*wx0
        bm.y = (m == ow) ? bw1 : 0.f;  // B[K1/K3, ow] = wy*wx1
        acc = __builtin_amdgcn_wmma_f32_16x16x4_f32(
            /*neg_a=*/false, a, /*neg_b=*/false, bm,
            /*c_mod=*/(short)0, acc, /*reuse_a=*/false, /*reuse_b=*/false);
    }

    // ---- store D: VGPR j holds M=j (lanes 0-15) / M=8+j (lanes 16-31), N=lane%16
    if (m < OW) {
        const int mbase = hi ? 8 : 0;
        float* op = out + (((size_t)n * NC + ct * CT + mbase) * OH + oh) * OW + m;
#pragma unroll
        for (int j = 0; j < 8; ++j)
            op[(size_t)j * (OH * OW)] = acc[j];
    }
}

extern "C" void kernel_launch(void* const* d_in, const int* in_sizes, int n_in,
                              void* d_out, int out_size, void* d_ws, size_t ws_size,
                              hipStream_t stream) {
    const float* fm   = (const float*)d_in[0];
    const float* rois = (const float*)d_in[1];
    float* out = (float*)d_out;
    (void)in_sizes; (void)n_in; (void)out_size; (void)d_ws; (void)ws_size;
    roi_align_wmma_kernel<<<NWAVES / WPB, BLOCK, 0, stream>>>(fm, rois, out);
}